// LS_solver_46608985096505
// MI455X (gfx1250) — compile-verified
//
#include <hip/hip_runtime.h>

// MoCap Kabsch solver: B=32768 frames, M=56 markers, J=24 joints.
// Memory-bound on Z (528 MB) -> stream Z once, stage X via TDM into LDS,
// per-thread Horn-quaternion rotation extraction (no SVD needed).

#define NB 32768
#define NM 56
#define NJ 24
#define FPB 8                 // frames per block
#define NT (FPB * NJ)         // 192 threads = 6 wave32

typedef unsigned int u32x4 __attribute__((ext_vector_type(4)));
typedef int          i32x8 __attribute__((ext_vector_type(8)));
typedef int          i32x4 __attribute__((ext_vector_type(4)));

#if defined(__gfx1250__) && __has_builtin(__builtin_amdgcn_tensor_load_to_lds)
#define HAVE_TDM 1
#else
#define HAVE_TDM 0
#endif

__global__ __launch_bounds__(NT) void mocap_kabsch(
    const float* __restrict__ X,   // [B,M,3]
    const float* __restrict__ Z,   // [B,M,J,3]
    const int*   __restrict__ W,   // [M,J] binary
    float* __restrict__ out)       // [B,J,3,4]
{
  __shared__ float sX[FPB * NM * 3];            // 1344 floats = 5376 B
  __shared__ unsigned long long jm[NJ];         // per-joint 56-bit marker masks

  const int tid = threadIdx.x;
  const int b0  = blockIdx.x * FPB;
  const int f   = tid / NJ;
  const int j   = tid - f * NJ;
  const int b   = b0 + f;

#if HAVE_TDM
  // --- Stage X[b0:b0+8, :, :] (contiguous 1344 f32) into LDS via TDM ---
  if (tid < 32) {                               // wave 0 issues one DMA
    const unsigned int nElem = (unsigned int)(FPB * NM * 3);   // 1344
    unsigned long long ga =
        (unsigned long long)(uintptr_t)(X + (size_t)b0 * (NM * 3));
    unsigned int ldsOff =
        (unsigned int)(unsigned long long)(uintptr_t)(&sX[0]); // low32 = LDS byte offset

    u32x4 g0;
    g0.x = 1u;                                  // count=1, user-mode, no gather
    g0.y = ldsOff;                              // lds_addr (bytes)
    g0.z = (unsigned int)(ga & 0xFFFFFFFFull);  // global_addr[31:0]
    g0.w = (unsigned int)((ga >> 32) & 0x01FFFFFFull) | 0x80000000u; // addr[56:32] | type=2

    i32x8 g1;
    g1[0] = 0x00020000;                         // workgroup_mask=0, data_size=4B
    g1[1] = (int)((nElem & 0xFFFFu) << 16);     // tensor_dim0[15:0]
    g1[2] = (int)(((nElem >> 16) & 0xFFFFu) | (1u << 16)); // tensor_dim0 hi | tensor_dim1=1
    g1[3] = (int)((nElem & 0xFFFFu) << 16);     // tile_dim0 = 1344
    g1[4] = 1;                                  // tile_dim1=1, tile_dim2=0
    g1[5] = (int)nElem;                         // tensor_dim0_stride lo32
    g1[6] = 0;
    g1[7] = 0;
    i32x4 gz4 = {0, 0, 0, 0};                   // groups 2/3 unused (<=2D tile)
    i32x8 gz8 = {0, 0, 0, 0, 0, 0, 0, 0};
    __builtin_amdgcn_tensor_load_to_lds(g0, g1, gz4, gz4, gz8, 0);
  }
#else
  {
    const float* xg = X + (size_t)b0 * (NM * 3);
    for (int i = tid; i < FPB * NM * 3; i += NT) sX[i] = xg[i];
  }
#endif

  // --- Pack per-joint marker masks (tiny, L2-resident) ---
  if (tid < NJ) {
    unsigned long long mk = 0ull;
    for (int m = 0; m < NM; ++m)
      mk |= (unsigned long long)(W[m * NJ + tid] & 1) << m;
    jm[tid] = mk;
  }

#if HAVE_TDM
  if (tid < 32) __builtin_amdgcn_s_wait_tensorcnt(0);
#endif
  __syncthreads();

  // --- Masked accumulation: S = sum w z x^T, sums of z, x, count ---
  const unsigned long long wm = jm[j];
  const float* zp = Z + ((size_t)b * (NM * NJ) + j) * 3;   // stride NJ*3 per marker
  const float* xp = sX + f * (NM * 3);

  float s00=0.f,s01=0.f,s02=0.f,s10=0.f,s11=0.f,s12=0.f,s20=0.f,s21=0.f,s22=0.f;
  float sz0=0.f,sz1=0.f,sz2=0.f, sx0=0.f,sx1=0.f,sx2=0.f, cnt=0.f;

  #pragma unroll 8
  for (int m = 0; m < NM; ++m) {
    float on = (float)((wm >> m) & 1ull);
    float za = zp[0] * on, zb = zp[1] * on, zc = zp[2] * on;
    float xa = xp[0], xb = xp[1], xc = xp[2];
    s00 = fmaf(za, xa, s00); s01 = fmaf(za, xb, s01); s02 = fmaf(za, xc, s02);
    s10 = fmaf(zb, xa, s10); s11 = fmaf(zb, xb, s11); s12 = fmaf(zb, xc, s12);
    s20 = fmaf(zc, xa, s20); s21 = fmaf(zc, xb, s21); s22 = fmaf(zc, xc, s22);
    sz0 += za; sz1 += zb; sz2 += zc;
    sx0 = fmaf(on, xa, sx0); sx1 = fmaf(on, xb, sx1); sx2 = fmaf(on, xc, sx2);
    cnt += on;
    zp += NJ * 3; xp += 3;
  }

  // --- Centered cross-covariance H = S - meanZ (sumX)^T ---
  const float inv = 1.0f / cnt;
  const float mz0 = sz0*inv, mz1 = sz1*inv, mz2 = sz2*inv;
  const float mx0 = sx0*inv, mx1 = sx1*inv, mx2 = sx2*inv;
  const float h00 = s00 - mz0*sx0, h01 = s01 - mz0*sx1, h02 = s02 - mz0*sx2;
  const float h10 = s10 - mz1*sx0, h11 = s11 - mz1*sx1, h12 = s12 - mz1*sx2;
  const float h20 = s20 - mz2*sx0, h21 = s21 - mz2*sx1, h22 = s22 - mz2*sx2;

  // --- Horn's 4x4 N matrix; max eigenvector = optimal proper rotation ---
  const float n00 =  h00 + h11 + h22;
  const float n01 =  h12 - h21;
  const float n02 =  h20 - h02;
  const float n03 =  h01 - h10;
  const float n11 =  h00 - h11 - h22;
  const float n12 =  h01 + h10;
  const float n13 =  h20 + h02;
  const float n22 = -h00 + h11 - h22;
  const float n23 =  h12 + h21;
  const float n33 = -h00 - h11 + h22;

  const float fr = n00*n00 + n11*n11 + n22*n22 + n33*n33 +
      2.0f * (n01*n01 + n02*n02 + n03*n03 + n12*n12 + n13*n13 + n23*n23);
  const float sig = sqrtf(fr) + 1e-12f;          // shift: N + sig*I is PSD
  const float a00 = n00 + sig, a11 = n11 + sig, a22 = n22 + sig, a33 = n33 + sig;

  float q0 = 1.0f, q1 = 0.02f, q2 = 0.04f, q3 = 0.06f;
  #pragma unroll
  for (int it = 0; it < 24; ++it) {
    float r0 = a00*q0 + n01*q1 + n02*q2 + n03*q3;
    float r1 = n01*q0 + a11*q1 + n12*q2 + n13*q3;
    float r2 = n02*q0 + n12*q1 + a22*q2 + n23*q3;
    float r3 = n03*q0 + n13*q1 + n23*q2 + a33*q3;
    float nn = r0*r0 + r1*r1 + r2*r2 + r3*r3;
    float ri = rsqrtf(nn + 1e-30f);
    q0 = r0*ri; q1 = r1*ri; q2 = r2*ri; q3 = r3*ri;
  }

  // --- Quaternion -> rotation (maps Z-frame into X-frame) ---
  const float xx = q1*q1, yy = q2*q2, zz2 = q3*q3;
  const float xy = q1*q2, xz = q1*q3, yz = q2*q3;
  const float wx = q0*q1, wy = q0*q2, wz = q0*q3;
  const float r00 = 1.0f - 2.0f*(yy + zz2);
  const float r01 = 2.0f*(xy - wz);
  const float r02 = 2.0f*(xz + wy);
  const float r10 = 2.0f*(xy + wz);
  const float r11 = 1.0f - 2.0f*(xx + zz2);
  const float r12 = 2.0f*(yz - wx);
  const float r20 = 2.0f*(xz - wy);
  const float r21 = 2.0f*(yz + wx);
  const float r22 = 1.0f - 2.0f*(xx + yy);

  const float t0 = mx0 - (r00*mz0 + r01*mz1 + r02*mz2);
  const float t1 = mx1 - (r10*mz0 + r11*mz1 + r12*mz2);
  const float t2 = mx2 - (r20*mz0 + r21*mz1 + r22*mz2);

  float4* o = reinterpret_cast<float4*>(out + ((size_t)b * NJ + j) * 12);
  o[0] = make_float4(r00, r01, r02, t0);
  o[1] = make_float4(r10, r11, r12, t1);
  o[2] = make_float4(r20, r21, r22, t2);
}

extern "C" void kernel_launch(void* const* d_in, const int* in_sizes, int n_in,
                              void* d_out, int out_size, void* d_ws, size_t ws_size,
                              hipStream_t stream) {
  (void)in_sizes; (void)n_in; (void)out_size; (void)d_ws; (void)ws_size;
  const float* X = (const float*)d_in[0];
  const float* Z = (const float*)d_in[1];
  const int*   W = (const int*)d_in[2];
  float* out = (float*)d_out;
  mocap_kabsch<<<dim3(NB / FPB), dim3(NT), 0, stream>>>(X, Z, W, out);
}